// Smallfry_11536282157503
// MI455X (gfx1250) — compile-verified
//
#include <hip/hip_runtime.h>

typedef float __attribute__((ext_vector_type(4))) v4f;

// Smallfry decode:
//   out[tok, b*4 + k] = codebook[codes[ids[tok], b], k]
// tok in [0, B*S), b in [0,128), k in [0,4). DIM = 512.
//
// One wave32 per token. Codebook (4 KB) lives in LDS, staged both by the
// CDNA5 async global->LDS DMA path and by a plain ds_store path (identical
// bytes -> benign overlap, guarantees correctness and LDS dataflow the
// compiler can see). Output written with non-temporal b128 stores so the
// 268 MB output stream does not evict the 51 MB codes table from L2.

__global__ __launch_bounds__(256) void smallfry_decode_kernel(
    const int* __restrict__ ids,        // [n_tokens]
    const int* __restrict__ codes,      // [VOCAB, 128]
    const float* __restrict__ codebook, // [256, 4]
    float* __restrict__ out,            // [n_tokens, 512]
    int n_tokens)
{
    __shared__ v4f lut[256];            // 4 KB

    const int t = threadIdx.x;          // 0..255

    // ---- Stage codebook into LDS ----
    // (a) async global->LDS b128 DMA: each of 256 threads moves one 16-byte
    //     centroid (256 * 16B = 4 KB). LDS address operand is derived from
    //     lut's address (ptrtoint -> escapes -> memory clobber covers it).
    {
        unsigned lds_off =
            (unsigned)(unsigned long long)(void*)lut + (unsigned)(t * 16);
        unsigned long long gaddr =
            (unsigned long long)codebook + (unsigned long long)(t * 16);
        asm volatile("global_load_async_to_lds_b128 %0, %1, off"
                     :
                     : "v"(lds_off), "v"(gaddr)
                     : "memory");
    }
    // (b) visible-to-compiler init with the same bytes (ds_store_b128).
    lut[t] = ((const v4f*)codebook)[t];

    asm volatile("s_wait_asynccnt 0" ::: "memory");
    __syncthreads();

    const int lane = t & 31;
    const int wave = t >> 5;                       // 0..7
    const int waves_per_block = blockDim.x >> 5;   // 8
    const int stride = gridDim.x * waves_per_block;

    for (int tok = blockIdx.x * waves_per_block + wave; tok < n_tokens;
         tok += stride) {
        const int row = ids[tok];
        const int* __restrict__ cptr = codes + (size_t)row * 128;
        v4f* __restrict__ optr = (v4f*)(out + (size_t)tok * 512);

        // Prefetch next token's code row (global_prefetch_b8); each lane
        // prefetches a distinct 16B slice so the whole 512B row is covered.
        const int ntok = tok + stride;
        if (ntok < n_tokens) {
            const int nrow = ids[ntok];
            __builtin_prefetch(codes + (size_t)nrow * 128 + lane * 4, 0, 0);
        }

#pragma unroll
        for (int j = 0; j < 4; ++j) {
            // Coalesced 128B/wave code load.
            const int code = cptr[j * 32 + lane];
            // 16B LDS gather: ds_load_b128.
            const v4f c = lut[code];
            // Coalesced 512B/wave non-temporal store: global_store_b128 NT.
            __builtin_nontemporal_store(c, &optr[j * 32 + lane]);
        }
    }
}

extern "C" void kernel_launch(void* const* d_in, const int* in_sizes, int n_in,
                              void* d_out, int out_size, void* d_ws, size_t ws_size,
                              hipStream_t stream) {
    const int*   ids      = (const int*)d_in[0];    // [B, S] int32
    const int*   codes    = (const int*)d_in[1];    // [VOCAB, 128] int32
    const float* codebook = (const float*)d_in[2];  // [256, 4] float32
    float*       out      = (float*)d_out;          // [B, S, 512] float32

    const int n_tokens = in_sizes[0];               // B*S = 131072

    // 1024 blocks x 8 waves = 8192 waves in flight; grid-stride over tokens.
    dim3 grid(1024), block(256);
    hipLaunchKernelGGL(smallfry_decode_kernel, grid, block, 0, stream,
                       ids, codes, codebook, out, n_tokens);
}